// QFBNLayer_44813688767074
// MI455X (gfx1250) — compile-verified
//
#include <hip/hip_runtime.h>
#include <hip/hip_bf16.h>
#include <cstdint>

// Problem constants (from reference): B=256, IN=1024, OUT=1024, K=3
static constexpr int B_DIM  = 256;
static constexpr int IN_DIM = 1024;
static constexpr int OUT_DIM = 1024;

static constexpr int BT  = 32;   // batch tile
static constexpr int OT  = 32;   // out tile
static constexpr int KT  = 64;   // in-dim chunk staged in LDS
static constexpr int KTP = 68;   // padded LDS row stride (68*4=272B, 16B aligned, bank-skewed)

// -------- gfx1250 hardware tanh (TRANS op v_tanh_f32) ----------------------
__device__ __forceinline__ float htanh(float x) {
#if defined(__gfx1250__)
#if __has_builtin(__builtin_amdgcn_tanhf)
  return __builtin_amdgcn_tanhf(x);
#else
  float r;
  asm volatile("v_tanh_f32 %0, %1" : "=v"(r) : "v"(x));
  return r;
#endif
#else
  return tanhf(x);
#endif
}

// -------- async global -> LDS copy (CDNA5, tracked by ASYNCcnt) ------------
__device__ __forceinline__ void async_copy_b128(const float* gsrc, uint32_t lds_off) {
#if defined(__gfx1250__)
  // GV mode: vdst = LDS byte address, vaddr = 64-bit global address, no saddr
  asm volatile("global_load_async_to_lds_b128 %0, %1, off"
               :: "v"(lds_off), "v"(gsrc) : "memory");
#endif
}

__device__ __forceinline__ void wait_async_all() {
#if defined(__gfx1250__)
  asm volatile("s_wait_asynccnt 0x0" ::: "memory");
#endif
}

__device__ __forceinline__ uint32_t lds_addr_u32(const void* p) {
  // generic pointer to LDS: low 32 bits are the LDS byte offset (ISA 10.2)
  return (uint32_t)(uintptr_t)p;
}

// ===========================================================================
// Kernel 1: W_eff[o,i] = softmax(f_logits[o,i,:]) . {w, tanh(w), sin(w)}
// Elementwise over 1M elements; ~20MB traffic, negligible vs kernel 2.
// ===========================================================================
__global__ __launch_bounds__(256)
void weff_kernel(const float* __restrict__ W,
                 const float* __restrict__ fl,
                 float* __restrict__ weff, int n) {
  int idx = blockIdx.x * 256 + threadIdx.x;
  if (idx >= n) return;
  float l0 = fl[3 * idx + 0];
  float l1 = fl[3 * idx + 1];
  float l2 = fl[3 * idx + 2];
  float m  = fmaxf(l0, fmaxf(l1, l2));
  float e0 = __expf(l0 - m);
  float e1 = __expf(l1 - m);
  float e2 = __expf(l2 - m);
  float inv = __frcp_rn(e0 + e1 + e2);
  float w = W[idx];
  float v = (e0 * w + e1 * htanh(w) + e2 * __sinf(w)) * inv;
  weff[idx] = v;
}

// ===========================================================================
// Kernel 2: out[b,o] = sum_i tanh(h[b,i] * W_eff[o,i]) + bias[o]
// 32x32 tile per 256-thread block, 2x2 outputs per thread, K chunks of 64
// double-buffered in LDS via async global->LDS b128 copies.
// ===========================================================================
__global__ __launch_bounds__(256)
void mix_tanh_sum_kernel(const float* __restrict__ h,
                         const float* __restrict__ weff,
                         const float* __restrict__ bias,
                         float* __restrict__ out) {
  __shared__ float hs[2][BT][KTP];
  __shared__ float ws[2][OT][KTP];

  const int tid   = threadIdx.x;
  const int bRow0 = blockIdx.y * BT;
  const int oCol0 = blockIdx.x * OT;
  const int tx = tid & 15;   // out sub-tile /2
  const int ty = tid >> 4;   // batch sub-tile /2

  // Issue async loads for one K-chunk into buffer `buf`.
  // Each tile is 32 rows x 64 floats = 512 x 16B quanta; 256 threads x 2 each.
  auto issue = [&](int buf, int chunk) {
    const int kbase = chunk * KT;
#pragma unroll
    for (int r = 0; r < 2; ++r) {
      const int q   = tid + r * 256;
      const int row = q >> 4;          // 0..31
      const int c4  = (q & 15) << 2;   // 0,4,...,60
      const float* gh = h    + (size_t)(bRow0 + row) * IN_DIM + kbase + c4;
      const float* gw = weff + (size_t)(oCol0 + row) * IN_DIM + kbase + c4;
      async_copy_b128(gh, lds_addr_u32(&hs[buf][row][c4]));
      async_copy_b128(gw, lds_addr_u32(&ws[buf][row][c4]));
    }
  };

  float a00 = 0.f, a01 = 0.f, a10 = 0.f, a11 = 0.f;

  issue(0, 0);
  wait_async_all();
  __syncthreads();

  constexpr int NCHUNK = IN_DIM / KT;  // 16
  for (int c = 0; c < NCHUNK; ++c) {
    const int buf = c & 1;
    if (c + 1 < NCHUNK) issue(buf ^ 1, c + 1);

    const float* hr0 = hs[buf][2 * ty + 0];
    const float* hr1 = hs[buf][2 * ty + 1];
    const float* wr0 = ws[buf][2 * tx + 0];
    const float* wr1 = ws[buf][2 * tx + 1];

#pragma unroll
    for (int k = 0; k < KT; k += 4) {
      const float4 av0 = *(const float4*)(hr0 + k);
      const float4 av1 = *(const float4*)(hr1 + k);
      const float4 wv0 = *(const float4*)(wr0 + k);
      const float4 wv1 = *(const float4*)(wr1 + k);

      a00 += htanh(av0.x * wv0.x);
      a01 += htanh(av0.x * wv1.x);
      a10 += htanh(av1.x * wv0.x);
      a11 += htanh(av1.x * wv1.x);

      a00 += htanh(av0.y * wv0.y);
      a01 += htanh(av0.y * wv1.y);
      a10 += htanh(av1.y * wv0.y);
      a11 += htanh(av1.y * wv1.y);

      a00 += htanh(av0.z * wv0.z);
      a01 += htanh(av0.z * wv1.z);
      a10 += htanh(av1.z * wv0.z);
      a11 += htanh(av1.z * wv1.z);

      a00 += htanh(av0.w * wv0.w);
      a01 += htanh(av0.w * wv1.w);
      a10 += htanh(av1.w * wv0.w);
      a11 += htanh(av1.w * wv1.w);
    }

    wait_async_all();   // next chunk's transfers finished (overlapped w/ compute)
    __syncthreads();
  }

  const int oc = oCol0 + 2 * tx;
  const float b0 = bias[oc + 0];
  const float b1 = bias[oc + 1];
  float* o0 = out + (size_t)(bRow0 + 2 * ty + 0) * OUT_DIM + oc;
  float* o1 = out + (size_t)(bRow0 + 2 * ty + 1) * OUT_DIM + oc;
  float2 r0; r0.x = a00 + b0; r0.y = a01 + b1;
  float2 r1; r1.x = a10 + b0; r1.y = a11 + b1;
  *(float2*)o0 = r0;
  *(float2*)o1 = r1;
}

// ===========================================================================
extern "C" void kernel_launch(void* const* d_in, const int* in_sizes, int n_in,
                              void* d_out, int out_size, void* d_ws, size_t ws_size,
                              hipStream_t stream) {
  const float* h  = (const float*)d_in[0];   // (256, 1024)
  const float* W  = (const float*)d_in[1];   // (1024, 1024)
  const float* b  = (const float*)d_in[2];   // (1024,)
  const float* fl = (const float*)d_in[3];   // (1024, 1024, 3)
  float* out = (float*)d_out;                // (256, 1024)
  float* weff = (float*)d_ws;                // (1024, 1024) scratch, 4MB

  const int n = in_sizes[1];                 // OUT*IN = 1M
  weff_kernel<<<(n + 255) / 256, 256, 0, stream>>>(W, fl, weff, n);

  dim3 grid(OUT_DIM / OT, B_DIM / BT);       // (32, 8)
  mix_tanh_sum_kernel<<<grid, 256, 0, stream>>>(h, weff, b, out);
}